// CellEmbedding_58076547776778
// MI455X (gfx1250) — compile-verified
//
#include <hip/hip_runtime.h>

typedef __attribute__((ext_vector_type(2))) float v2f;
typedef __attribute__((ext_vector_type(8))) float v8f;

#define IN_FEAT  256
#define OUT_FEAT 64
#define KPAIRS   (IN_FEAT / 2)   // 128 K-pairs
#define EPG      128             // edges per scatter group

// ---------------------------------------------------------------- zero out
__global__ void gcn_zero_f32(float* __restrict__ p, int n) {
    int i = blockIdx.x * blockDim.x + threadIdx.x;
    if (i < n) p[i] = 0.0f;
}

// ---------------------------------------------- proj = feat @ W, fp32 WMMA
// One wave computes a 16-row x 64-col output tile as four 16x16 accumulators.
// K advances in steps of 4 using V_WMMA_F32_16X16X4_F32.
//
// A 16x4 f32 fragment (ISA 7.12.2): lane m = lane&15 holds row m;
//   VGPR0 = K = 2*(lane>>4), VGPR1 = K = 2*(lane>>4)+1  -> one aligned v2f load.
// B 4x16 f32 fragment: VGPR j holds row K = j + 2*(lane>>4), N = lane&15.
//   Weight is staged in LDS interleaved as K-pairs, lW[kp][n] = {W[2kp][n],
//   W[2kp+1][n]}, so each B fragment is ONE aligned ds_load_b64 into a
//   consecutive even VGPR pair (no assembly movs).
// C/D 16x16 f32: VGPR v holds M = v + 8*(lane>>4), N = lane&15.
__global__ void __launch_bounds__(128) gcn_gemm_wmma(
    const float* __restrict__ feat, const float* __restrict__ W,
    float* __restrict__ proj, int n_nodes)
{
    __shared__ v2f lW[KPAIRS * OUT_FEAT];             // 64 KB, K-pair interleaved
    for (int i = threadIdx.x; i < KPAIRS * OUT_FEAT; i += 128) {
        const int kp = i >> 6;                        // K pair index
        const int n  = i & 63;                        // output column
        v2f w;
        w.x = W[(2 * kp + 0) * OUT_FEAT + n];
        w.y = W[(2 * kp + 1) * OUT_FEAT + n];
        lW[i] = w;
    }
    __syncthreads();

    const int wave = threadIdx.x >> 5;                // 0..3
    const int lane = threadIdx.x & 31;
    const int m    = lane & 15;
    const int half = lane >> 4;                       // 0/1

    const int row_base = (blockIdx.x * 4 + wave) * 16;
    if (row_base >= n_nodes) return;                  // wave-uniform

    // A: even element offset -> 8-byte aligned v2f loads
    const v2f* arow = (const v2f*)(feat + (size_t)(row_base + m) * IN_FEAT) + half;

    v8f c0 = {}, c1 = {}, c2 = {}, c3 = {};

    for (int k = 0; k < IN_FEAT; k += 4) {
        const v2f a = arow[k >> 1];                   // K = k+2*half, k+2*half+1

        const v2f* wb = lW + ((k >> 1) + half) * OUT_FEAT + m;
        const v2f b0 = wb[0];
        const v2f b1 = wb[16];
        const v2f b2 = wb[32];
        const v2f b3 = wb[48];

        c0 = __builtin_amdgcn_wmma_f32_16x16x4_f32(false, a, false, b0, (short)0, c0, false, false);
        c1 = __builtin_amdgcn_wmma_f32_16x16x4_f32(false, a, false, b1, (short)0, c1, false, false);
        c2 = __builtin_amdgcn_wmma_f32_16x16x4_f32(false, a, false, b2, (short)0, c2, false, false);
        c3 = __builtin_amdgcn_wmma_f32_16x16x4_f32(false, a, false, b3, (short)0, c3, false, false);
    }

    float* prow = proj + (size_t)row_base * OUT_FEAT + m;
#pragma unroll
    for (int v = 0; v < 8; ++v) {
        const size_t r = (size_t)(v + 8 * half) * OUT_FEAT;
        prow[r + 0]  = c0[v];
        prow[r + 16] = c1[v];
        prow[r + 32] = c2[v];
        prow[r + 48] = c3[v];
    }
}

// ------------------------- out[row] += val * proj[col], sorted-row scatter
// blockDim = 256 = 4 groups x 64 feature lanes. Each group walks EPG
// contiguous edges; rows are sorted, so accumulate in a register while the
// row is unchanged and flush once per run with atomicAdd (L2-resident).
__global__ void __launch_bounds__(256) gcn_scatter(
    const float* __restrict__ proj,
    const int*   __restrict__ rows,
    const int*   __restrict__ cols,
    const float* __restrict__ vals,
    float* __restrict__ out, int n_edges)
{
    const int f     = threadIdx.x & 63;               // feature column
    const int sub   = threadIdx.x >> 6;               // 0..3
    const int group = blockIdx.x * 4 + sub;

    int e0 = group * EPG;
    if (e0 >= n_edges) return;
    int e1 = e0 + EPG;
    if (e1 > n_edges) e1 = n_edges;

    float acc = 0.0f;
    int cur_row = rows[e0];

    for (int e = e0; e < e1; ++e) {
        int r = rows[e];
        if (r != cur_row) {
            atomicAdd(&out[(size_t)cur_row * OUT_FEAT + f], acc);
            acc = 0.0f;
            cur_row = r;
        }
        int c = cols[e];
        acc += vals[e] * proj[(size_t)c * OUT_FEAT + f];
    }
    atomicAdd(&out[(size_t)cur_row * OUT_FEAT + f], acc);
}

extern "C" void kernel_launch(void* const* d_in, const int* in_sizes, int n_in,
                              void* d_out, int out_size, void* d_ws, size_t ws_size,
                              hipStream_t stream) {
    const float* feat = (const float*)d_in[0];
    const float* W    = (const float*)d_in[1];
    const int*   rows = (const int*)d_in[2];
    const int*   cols = (const int*)d_in[3];
    const float* vals = (const float*)d_in[4];
    float* out  = (float*)d_out;
    float* proj = (float*)d_ws;                       // [n_nodes, 64] fp32

    const int n_nodes = in_sizes[0] / IN_FEAT;
    const int n_edges = in_sizes[2];

    // 1) zero the (poisoned) output
    const int zn = n_nodes * OUT_FEAT;
    gcn_zero_f32<<<(zn + 255) / 256, 256, 0, stream>>>(out, zn);

    // 2) dense projection via fp32 WMMA
    const int row_tiles = (n_nodes + 15) / 16;
    gcn_gemm_wmma<<<(row_tiles + 3) / 4, 128, 0, stream>>>(feat, W, proj, n_nodes);

    // 3) gather + weighted segment scatter-add
    const int groups = (n_edges + EPG - 1) / EPG;
    gcn_scatter<<<(groups + 3) / 4, 256, 0, stream>>>(proj, rows, cols, vals, out, n_edges);
}